// GNNClassifier_64527588655724
// MI455X (gfx1250) — compile-verified
//
#include <hip/hip_runtime.h>
#include <hip/hip_bf16.h>

typedef float v2f __attribute__((ext_vector_type(2)));
typedef float v8f __attribute__((ext_vector_type(8)));

// ---------------------------------------------------------------------------
// Utility fills
// ---------------------------------------------------------------------------
__global__ void fill_zero_f32(float* __restrict__ p, long long n) {
    long long t = (long long)blockIdx.x * blockDim.x + threadIdx.x;
    if (t < n) p[t] = 0.0f;
}

__global__ void fill_zero_u32(unsigned int* __restrict__ p, int n) {
    int t = blockIdx.x * blockDim.x + threadIdx.x;
    if (t < n) p[t] = 0u;
}

// ---------------------------------------------------------------------------
// Degree / normalization
// ---------------------------------------------------------------------------
__global__ void edge_degree(const int* __restrict__ dst,
                            unsigned int* __restrict__ cnt, int E) {
    int e = blockIdx.x * blockDim.x + threadIdx.x;
    if (e < E) atomicAdd(&cnt[dst[e]], 1u);
}

__global__ void compute_dinv(const unsigned int* __restrict__ cnt,
                             float* __restrict__ dinv, int N) {
    int i = blockIdx.x * blockDim.x + threadIdx.x;
    if (i < N) dinv[i] = rsqrtf((float)cnt[i] + 1.0f);  // +1 self loop
}

// ---------------------------------------------------------------------------
// WMMA GEMM: C[nrows, COLS] = A[nrows, 128] * W[128, COLS]   (fp32 WMMA)
// Block: 256 threads = 8 waves; block tile = 128 rows x COLS cols.
// Each wave owns a 16-row strip and all COLS/16 column tiles.
// K staged through LDS in chunks of 16.
// ---------------------------------------------------------------------------
template <int COLS>
__global__ __launch_bounds__(256)
void gemm_wmma(const float* __restrict__ A, const float* __restrict__ Wm,
               float* __restrict__ C, int nrows) {
    constexpr int K = 128;
    constexpr int KC = 16;               // K chunk staged in LDS
    constexpr int TILES = COLS / 16;
    constexpr int XS_STRIDE = 17;        // pad to avoid LDS bank conflicts

    __shared__ float Xs[128 * XS_STRIDE];  // [row][k within chunk]
    __shared__ float Ws[KC * COLS];        // [k within chunk][col]

    const int tid  = threadIdx.x;
    const int wave = tid >> 5;
    const int lane = tid & 31;
    const int half = lane >> 4;   // 0: lanes 0-15, 1: lanes 16-31
    const int l16  = lane & 15;
    const int blockRow = blockIdx.x * 128;

    v8f acc[TILES];
    #pragma unroll
    for (int t = 0; t < TILES; ++t)
        #pragma unroll
        for (int r = 0; r < 8; ++r) acc[t][r] = 0.0f;

    for (int kc = 0; kc < K; kc += KC) {
        // Stage A tile: 128 rows x 16 k-cols (coalesced 16-float runs)
        for (int e = tid; e < 128 * KC; e += 256) {
            int r = e >> 4, c = e & 15;
            int row = blockRow + r;
            if (row >= nrows) row = 0;   // safe clamp; result never stored
            Xs[r * XS_STRIDE + c] = A[(long long)row * K + kc + c];
        }
        // Stage W chunk: 16 x COLS (fully coalesced)
        for (int e = tid; e < KC * COLS; e += 256) {
            int r = e / COLS, c = e % COLS;   // COLS is power of two
            Ws[r * COLS + c] = Wm[(long long)(kc + r) * COLS + c];
        }
        __syncthreads();

        #pragma unroll
        for (int kk = 0; kk < KC; kk += 4) {
            // A fragment (16x4 f32): lanes0-15 hold K=kk,kk+1; lanes16-31 K=kk+2,kk+3
            const int k0 = kk + 2 * half;
            v2f a;
            a.x = Xs[(wave * 16 + l16) * XS_STRIDE + k0];
            a.y = Xs[(wave * 16 + l16) * XS_STRIDE + k0 + 1];
            #pragma unroll
            for (int t = 0; t < TILES; ++t) {
                // B fragment (4x16 f32): VGPR0 rows {kk, kk+2}, VGPR1 rows {kk+1, kk+3}
                v2f b;
                b.x = Ws[(kk + 2 * half) * COLS + t * 16 + l16];
                b.y = Ws[(kk + 2 * half + 1) * COLS + t * 16 + l16];
                acc[t] = __builtin_amdgcn_wmma_f32_16x16x4_f32(
                    /*neg_a=*/false, a, /*neg_b=*/false, b,
                    /*c_mod=*/(short)0, acc[t],
                    /*reuse_a=*/false, /*reuse_b=*/false);
            }
        }
        __syncthreads();
    }

    // C/D layout: VGPR r -> (M = r + 8*half, N = l16) within the 16x16 tile
    const int rbase = blockRow + wave * 16;
    #pragma unroll
    for (int r = 0; r < 8; ++r) {
        int row = rbase + r + 8 * half;
        if (row < nrows) {
            #pragma unroll
            for (int t = 0; t < TILES; ++t)
                C[(long long)row * COLS + t * 16 + l16] = acc[t][r];
        }
    }
}

// ---------------------------------------------------------------------------
// Edge scatter-add:  agg[dst] += h[src] * dinv[src]*dinv[dst]
// One lane per (edge, feature) -> wave reads a contiguous feature strip of h,
// issues native global_atomic_add_f32 per lane.
// ---------------------------------------------------------------------------
template <int F>
__global__ __launch_bounds__(256)
void edge_aggregate(const float* __restrict__ h, const int* __restrict__ src,
                    const int* __restrict__ dst, const float* __restrict__ dinv,
                    float* __restrict__ agg, long long total) {
    constexpr int LOG2F = (F == 128) ? 7 : 5;
    long long t = (long long)blockIdx.x * blockDim.x + threadIdx.x;
    if (t >= total) return;
    int e = (int)(t >> LOG2F);
    int f = (int)(t & (F - 1));
    int s = src[e];
    int d = dst[e];
    float w = dinv[s] * dinv[d];
    float v = h[(long long)s * F + f] * w;
    unsafeAtomicAdd(&agg[(long long)d * F + f], v);
}

// ---------------------------------------------------------------------------
// z = relu(agg + dinv^2 * h + b), written in-place over agg
// ---------------------------------------------------------------------------
template <int F>
__global__ __launch_bounds__(256)
void add_self_bias_relu(float* __restrict__ agg, const float* __restrict__ h,
                        const float* __restrict__ dinv, const float* __restrict__ b,
                        int N) {
    constexpr int LOG2F = (F == 128) ? 7 : 5;
    long long t = (long long)blockIdx.x * blockDim.x + threadIdx.x;
    if (t >= (long long)N * F) return;
    int i = (int)(t >> LOG2F);
    int f = (int)(t & (F - 1));
    float di = dinv[i];
    float v = agg[t] + di * di * h[t] + b[f];
    agg[t] = fmaxf(v, 0.0f);
}

// ---------------------------------------------------------------------------
// MLP head: logits = (relu(z2[q] @ Wm1 + bm1) @ Wm2 + bm2)
// One thread per query (4096 total; trivial compute, weights are SGPR-uniform)
// ---------------------------------------------------------------------------
__global__ __launch_bounds__(256)
void mlp_head(const float* __restrict__ z2, const int* __restrict__ qidx,
              const float* __restrict__ Wm1, const float* __restrict__ bm1,
              const float* __restrict__ Wm2, const float* __restrict__ bm2,
              float* __restrict__ out, int Q) {
    int i = blockIdx.x * blockDim.x + threadIdx.x;
    if (i >= Q) return;
    int node = qidx[i];
    float in[32];
    #pragma unroll
    for (int f = 0; f < 32; ++f) in[f] = z2[(long long)node * 32 + f];
    float accum = bm2[0];
    #pragma unroll 4
    for (int j = 0; j < 64; ++j) {
        float hj = bm1[j];
        #pragma unroll
        for (int f = 0; f < 32; ++f) hj = fmaf(in[f], Wm1[f * 64 + j], hj);
        hj = fmaxf(hj, 0.0f);
        accum = fmaf(hj, Wm2[j], accum);
    }
    out[i] = accum;
}

// ---------------------------------------------------------------------------
// Launch
// ---------------------------------------------------------------------------
static inline unsigned cdivu(long long a, long long b) {
    return (unsigned)((a + b - 1) / b);
}

extern "C" void kernel_launch(void* const* d_in, const int* in_sizes, int n_in,
                              void* d_out, int out_size, void* d_ws, size_t ws_size,
                              hipStream_t stream) {
    const float* x    = (const float*)d_in[0];
    const int*   ei   = (const int*)d_in[1];
    const int*   qidx = (const int*)d_in[2];
    const float* W1   = (const float*)d_in[3];
    const float* b1   = (const float*)d_in[4];
    const float* W2   = (const float*)d_in[5];
    const float* b2   = (const float*)d_in[6];
    const float* Wm1  = (const float*)d_in[7];
    const float* bm1  = (const float*)d_in[8];
    const float* Wm2  = (const float*)d_in[9];
    const float* bm2  = (const float*)d_in[10];

    const int N = in_sizes[0] / 128;      // 100000
    const int E = in_sizes[1] / 2;        // 1.6M
    const int Q = in_sizes[2];            // 4096
    const int* src = ei;
    const int* dst = ei + E;

    // Workspace carving (256B aligned)
    char* ws = (char*)d_ws;
    size_t off = 0;
    auto carve = [&](size_t bytes) -> void* {
        void* p = ws + off;
        off += (bytes + 255) & ~(size_t)255;
        return p;
    };
    float*        h1   = (float*)carve((size_t)N * 128 * sizeof(float));
    float*        agg1 = (float*)carve((size_t)N * 128 * sizeof(float)); // becomes z1 in-place
    float*        h2   = (float*)carve((size_t)N * 32 * sizeof(float));
    float*        agg2 = (float*)carve((size_t)N * 32 * sizeof(float));  // becomes z2 in-place
    unsigned int* cnt  = (unsigned int*)carve((size_t)N * sizeof(unsigned int));
    float*        dinv = (float*)carve((size_t)N * sizeof(float));

    // --- degrees + symmetric norm ---
    fill_zero_u32<<<cdivu(N, 256), 256, 0, stream>>>(cnt, N);
    edge_degree<<<cdivu(E, 256), 256, 0, stream>>>(dst, cnt, E);
    compute_dinv<<<cdivu(N, 256), 256, 0, stream>>>(cnt, dinv, N);

    // --- layer 1 ---
    gemm_wmma<128><<<cdivu(N, 128), 256, 0, stream>>>(x, W1, h1, N);
    fill_zero_f32<<<cdivu((long long)N * 128, 256), 256, 0, stream>>>(agg1, (long long)N * 128);
    edge_aggregate<128><<<cdivu((long long)E * 128, 256), 256, 0, stream>>>(
        h1, src, dst, dinv, agg1, (long long)E * 128);
    add_self_bias_relu<128><<<cdivu((long long)N * 128, 256), 256, 0, stream>>>(
        agg1, h1, dinv, b1, N);

    // --- layer 2 ---
    gemm_wmma<32><<<cdivu(N, 128), 256, 0, stream>>>(agg1, W2, h2, N);
    fill_zero_f32<<<cdivu((long long)N * 32, 256), 256, 0, stream>>>(agg2, (long long)N * 32);
    edge_aggregate<32><<<cdivu((long long)E * 32, 256), 256, 0, stream>>>(
        h2, src, dst, dinv, agg2, (long long)E * 32);
    add_self_bias_relu<32><<<cdivu((long long)N * 32, 256), 256, 0, stream>>>(
        agg2, h2, dinv, b2, N);

    // --- MLP head ---
    mlp_head<<<cdivu(Q, 256), 256, 0, stream>>>(agg2, qidx, Wm1, bm1, Wm2, bm2,
                                                (float*)d_out, Q);
}